// VectorQuantizer_78048145703117
// MI455X (gfx1250) — compile-verified
//
#include <hip/hip_runtime.h>
#include <hip/hip_bf16.h>

typedef float v2f __attribute__((ext_vector_type(2)));
typedef float v8f __attribute__((ext_vector_type(8)));
typedef unsigned int u32x4 __attribute__((ext_vector_type(4)));
typedef int i32x8 __attribute__((ext_vector_type(8)));
typedef int i32x4 __attribute__((ext_vector_type(4)));

#define N_E       8192
#define E_DIM     64
#define BETA      0.25f
#define N_TOKENS  32768

#define CHUNK       256   // codebook entries staged in LDS per iteration
#define LDS_STRIDE  68    // 64 + 4 pad floats -> conflict-free b64 B-fragment reads
#define WAVES_PER_WG 8
#define ROWS_PER_WG (WAVES_PER_WG * 16)   // 128 tokens per workgroup

#if defined(__gfx1250__) && __has_builtin(__builtin_amdgcn_tensor_load_to_lds)
#define HAVE_TDM 1
#else
#define HAVE_TDM 0
#endif

__global__ void vq_init_loss(float* loss) { *loss = 0.0f; }

#if HAVE_TDM
// Issue one TDM 2D tile load: CHUNK x E_DIM fp32, codebook row-major,
// tensor_dim1 = rows remaining (hardware zero-fills OOB rows),
// LDS padding 4 dwords after every 64 dwords -> LDS_STRIDE=68 layout.
__device__ __forceinline__ void tdm_load_chunk(const float* cb, int cbase, int end,
                                               unsigned lds_byte_addr)
{
    unsigned long long ga = (unsigned long long)(const void*)(cb + (size_t)cbase * E_DIM);
    unsigned rows_left = (unsigned)(end - cbase);

    u32x4 g0;
    g0[0] = 1u;                                             // count=1 (valid user D#)
    g0[1] = lds_byte_addr;                                  // lds_addr [63:32]
    g0[2] = (unsigned)(ga & 0xFFFFFFFFu);                   // global_addr [95:64]
    g0[3] = (unsigned)((ga >> 32) & 0x01FFFFFFu)            // global_addr [120:96]
          | (2u << 30);                                     // type=2 ("image") [127:126]

    i32x8 g1;
    g1[0] = (int)((2u << 16)       // data_size = 4 bytes
                | (1u << 20)       // pad_enable
                | (5u << 22)       // pad_interval: 64 dwords (1<<5 8-byte units)
                | (3u << 25));     // pad_amount: 4 dwords
    g1[1] = (int)(((unsigned)E_DIM & 0xFFFFu) << 16);       // tensor_dim0[15:0] @ [79:48]
    g1[2] = (int)((rows_left & 0xFFFFu) << 16);             // tensor_dim1[15:0] @ [111:80]
    g1[3] = (int)(((rows_left >> 16) & 0xFFFFu)             // tensor_dim1[31:16]
                | ((unsigned)E_DIM << 16));                 // tile_dim0 = 64 @ [127:112]
    g1[4] = (int)(CHUNK & 0xFFFFu);                         // tile_dim1 = 256; tile_dim2 = 0
    g1[5] = E_DIM;                                          // tensor_dim0_stride[31:0] = 64
    g1[6] = 0;                                              // stride0 hi | stride1 lo (2D: unused)
    g1[7] = 0;

    i32x4 gz4 = {0, 0, 0, 0};
    i32x8 gz8 = {0, 0, 0, 0, 0, 0, 0, 0};
    // amdgpu-toolchain (clang-23) 6-arg form: (g0, g1, g2, g3, g4, cpol)
    __builtin_amdgcn_tensor_load_to_lds(g0, g1, gz4, gz4, gz8, 0);
}
#endif

__global__ __launch_bounds__(256)
void vq_wmma_kernel(const float* __restrict__ x,
                    const float* __restrict__ cb,
                    const int*   __restrict__ startp,
                    const int*   __restrict__ endp,
                    float* __restrict__ xq_out,
                    float* __restrict__ loss_out,
                    float* __restrict__ idx_out)
{
    __shared__ float lds_cb[2][CHUNK * LDS_STRIDE];   // double-buffered codebook tiles
    __shared__ float lds_cn[2][CHUNK];                // per-entry squared norms

    const int start = startp[0];
    const int end   = endp[0];
    const int nchunks = (end - start + CHUNK - 1) / CHUNK;

    const int tid   = threadIdx.x;
    const int wave  = tid >> 5;
    const int lane  = tid & 31;
    const int nl    = lane & 15;          // M (for A/C) or N (for B/C) within tile
    const int khalf = (lane >> 4) * 2;    // K sub-offset within a 4-wide K step
    const int row_base = blockIdx.x * ROWS_PER_WG + wave * 16;

    // ---- Load A fragment: 16x64 fp32 tokens in WMMA 16x16x4 A layout ----
    v2f a[16];
    {
        const float* xr = x + (size_t)(row_base + nl) * E_DIM;
        #pragma unroll
        for (int s = 0; s < 16; ++s)
            a[s] = *(const v2f*)(xr + 4 * s + khalf);
    }

    float best_d[8];
    int   best_i[8];
    #pragma unroll
    for (int v = 0; v < 8; ++v) { best_d[v] = 3.4e38f; best_i[v] = 0; }

#if HAVE_TDM
    if (wave == 0)   // one TDM issue per workgroup (wave-uniform branch)
        tdm_load_chunk(cb, start, end, (unsigned)(size_t)(void*)&lds_cb[0][0]);
#endif

    for (int ci = 0; ci < nchunks; ++ci) {
        const int cbase = start + ci * CHUNK;
        const int buf   = ci & 1;

#if HAVE_TDM
        if (wave == 0) {
            if (ci + 1 < nchunks) {
                tdm_load_chunk(cb, cbase + CHUNK, end,
                               (unsigned)(size_t)(void*)&lds_cb[buf ^ 1][0]);
                __builtin_amdgcn_s_wait_tensorcnt(1);  // in-order: chunk ci has landed
            } else {
                __builtin_amdgcn_s_wait_tensorcnt(0);
            }
        }
#else
        // Fallback: synchronous cooperative staging (padded layout, zero-fill tail)
        #pragma unroll
        for (int i = 0; i < (CHUNK * E_DIM) / (256 * 4); ++i) {
            int q = (tid + i * 256) * 4;
            int e = q >> 6;
            int d = q & 63;
            float4 val = make_float4(0.f, 0.f, 0.f, 0.f);
            if (cbase + e < end)
                val = *(const float4*)(cb + (size_t)(cbase + e) * E_DIM + d);
            *(float4*)(&lds_cb[buf][e * LDS_STRIDE + d]) = val;
        }
#endif
        __syncthreads();   // buffer `buf` ready for all waves

        // ---- Per-entry squared norms (one entry per thread, float4 LDS reads) ----
        {
            float s = 0.f;
            const float4* p = (const float4*)&lds_cb[buf][tid * LDS_STRIDE];
            #pragma unroll
            for (int d = 0; d < E_DIM / 4; ++d) {
                float4 v = p[d];
                s = fmaf(v.x, v.x, s); s = fmaf(v.y, v.y, s);
                s = fmaf(v.z, v.z, s); s = fmaf(v.w, v.w, s);
            }
            lds_cn[buf][tid] = (cbase + tid < end) ? s : 3.4e38f;
        }
        __syncthreads();

        // ---- 16 column tiles of 16 codebook entries each ----
        for (int t = 0; t < CHUNK / 16; ++t) {
            const float* bp = &lds_cb[buf][(t * 16 + nl) * LDS_STRIDE];
            v8f c = {0.f, 0.f, 0.f, 0.f, 0.f, 0.f, 0.f, 0.f};
            #pragma unroll
            for (int s = 0; s < 16; ++s) {
                v2f b = *(const v2f*)(bp + 4 * s + khalf);  // B fragment (K x N)
                c = __builtin_amdgcn_wmma_f32_16x16x4_f32(
                        /*neg_a=*/false, a[s], /*neg_b=*/false, b,
                        /*c_mod=*/(short)0, c, /*reuse_a=*/false, /*reuse_b=*/false);
            }
            float cn  = lds_cn[buf][t * 16 + nl];
            int   col = cbase + t * 16 + nl;
            #pragma unroll
            for (int v = 0; v < 8; ++v) {
                // distance up to the per-row constant ||x||^2; col strictly increases
                // per lane slot, so strict `<` == first-index argmin semantics
                float d = fmaf(-2.0f, c[v], cn);
                if (d < best_d[v]) { best_d[v] = d; best_i[v] = col; }
            }
        }
        __syncthreads();   // all waves done with `buf` before it is refilled
    }

    // ---- Cross-lane argmin within each 16-lane half ----
    // (C layout: lanes 0-15 hold rows 0-7, lanes 16-31 hold rows 8-15)
    #pragma unroll
    for (int v = 0; v < 8; ++v) {
        #pragma unroll
        for (int off = 8; off >= 1; off >>= 1) {
            float od = __shfl_xor(best_d[v], off, 32);
            int   oi = __shfl_xor(best_i[v], off, 32);
            if (od < best_d[v] || (od == best_d[v] && oi < best_i[v])) {
                best_d[v] = od; best_i[v] = oi;
            }
        }
    }

    // ---- Write indices (as float, per harness output dtype) ----
    if (nl == 0) {  // lane 0 -> rows 0..7, lane 16 -> rows 8..15
        #pragma unroll
        for (int v = 0; v < 8; ++v)
            idx_out[row_base + (lane >> 4) * 8 + v] = (float)best_i[v];
    }

    // ---- Gather x_q = codebook[idx] + fused loss partial ----
    float lsum = 0.f;
    #pragma unroll
    for (int r = 0; r < 16; ++r) {
        int idx = __shfl(best_i[r & 7], (r < 8) ? 0 : 16, 32);
        const float2 cq = *(const float2*)(cb + (size_t)idx * E_DIM + 2 * lane);
        size_t o = (size_t)(row_base + r) * E_DIM + 2 * lane;
        const float2 xv = *(const float2*)(x + o);
        *(float2*)(xq_out + o) = cq;
        float dx = cq.x - xv.x, dy = cq.y - xv.y;
        lsum = fmaf(dx, dx, lsum);
        lsum = fmaf(dy, dy, lsum);
    }
    #pragma unroll
    for (int off = 16; off >= 1; off >>= 1)
        lsum += __shfl_xor(lsum, off, 32);
    if (lane == 0)
        atomicAdd(loss_out, lsum * ((1.0f + BETA) / ((float)N_TOKENS * (float)E_DIM)));
}

extern "C" void kernel_launch(void* const* d_in, const int* in_sizes, int n_in,
                              void* d_out, int out_size, void* d_ws, size_t ws_size,
                              hipStream_t stream) {
    const float* x     = (const float*)d_in[0];
    const float* cb    = (const float*)d_in[1];
    const int*   start = (const int*)d_in[2];
    const int*   end   = (const int*)d_in[3];
    // d_in[4] = use_sk (0 in reference path) -> ignored

    float* out  = (float*)d_out;
    float* xq   = out;                                  // N_TOKENS * E_DIM
    float* loss = out + (size_t)N_TOKENS * E_DIM;       // 1
    float* idx  = loss + 1;                             // N_TOKENS

    vq_init_loss<<<1, 1, 0, stream>>>(loss);
    vq_wmma_kernel<<<N_TOKENS / ROWS_PER_WG, 256, 0, stream>>>(
        x, cb, start, end, xq, loss, idx);
}